// AggrGATGated_52905407152679
// MI455X (gfx1250) — compile-verified
//
#include <hip/hip_runtime.h>
#include <math.h>

// Problem constants (from reference): B=2, N=50000, F=256, U=256, FE=128, T=3, E=131072
#define BN      100000   // B*N
#define FDIM    256
#define UDIM    256
#define FEDIM   128
#define NTYPE   3
#define NEDGE   131072

typedef __attribute__((ext_vector_type(2))) float v2f;
typedef __attribute__((ext_vector_type(8))) float v8f;

// ---------------------------------------------------------------------------
// Kernel 0: zero the output accumulator (harness poisons d_out with 0xAA).
// ---------------------------------------------------------------------------
__global__ __launch_bounds__(256) void zero_f32(float* __restrict__ p, int n4) {
    int i = blockIdx.x * blockDim.x + threadIdx.x;
    int stride = gridDim.x * blockDim.x;
    float4 z = make_float4(0.f, 0.f, 0.f, 0.f);
    for (; i < n4; i += stride) ((float4*)p)[i] = z;
}

// ---------------------------------------------------------------------------
// Kernel 1: src_gated[BN,U] = features[BN,F] @ W_gate[F,U]   (exact fp32 WMMA)
// Block = 256 threads = 8 waves. Block computes a 16-row x 256-col tile.
// Wave w computes cols [w*32, w*32+32) as two 16x16 WMMA accumulators.
// ---------------------------------------------------------------------------
#define APITCH 260  // 256 + 4 pad: row stride 1040B -> 16 rows hit 16 distinct banks
__global__ __launch_bounds__(256) void gate_gemm(
    const float* __restrict__ feat,   // [BN, F]
    const float* __restrict__ Wg,     // [F, U]
    float* __restrict__ dst)          // [BN, U]
{
    __shared__ float As[16 * APITCH];

    const int row0 = blockIdx.x * 16;

    // Cooperative load of 16x256 fp32 A tile (1024 float4's, 4 per thread)
    for (int i = threadIdx.x; i < 16 * (FDIM / 4); i += 256) {
        int r  = i >> 6;        // FDIM/4 = 64 float4 per row
        int c4 = i & 63;
        float4 v = ((const float4*)(feat + (size_t)(row0 + r) * FDIM))[c4];
        float* s = &As[r * APITCH + c4 * 4];
        s[0] = v.x; s[1] = v.y; s[2] = v.z; s[3] = v.w;
    }
    __syncthreads();

    const int lane = threadIdx.x & 31;
    const int wave = threadIdx.x >> 5;
    const int m    = lane & 15;       // A row / C column index
    const int kh   = lane >> 4;       // K half-select (lanes 16-31 hold K+2,K+3)
    const int n0   = wave * 32 + m;   // first output column for this lane

    v8f acc0 = {};
    v8f acc1 = {};

    for (int k = 0; k < FDIM; k += 4) {
        const int kk = k + 2 * kh;
        // A fragment: A[m][kk], A[m][kk+1]  (float2, 8B aligned: kk is even)
        v2f a = *(const v2f*)&As[m * APITCH + kk];
        // B fragments: B[kk][n], B[kk+1][n] for the two 16-col tiles
        v2f b0, b1;
        b0.x = Wg[(size_t)kk * UDIM + n0];
        b0.y = Wg[(size_t)(kk + 1) * UDIM + n0];
        b1.x = Wg[(size_t)kk * UDIM + n0 + 16];
        b1.y = Wg[(size_t)(kk + 1) * UDIM + n0 + 16];
        acc0 = __builtin_amdgcn_wmma_f32_16x16x4_f32(false, a, false, b0,
                                                     (short)0, acc0, false, false);
        acc1 = __builtin_amdgcn_wmma_f32_16x16x4_f32(false, a, false, b1,
                                                     (short)0, acc1, false, false);
    }

    // C layout: lanes 0-15 -> M = v, lanes 16-31 -> M = v + 8; N = n0
    const int mr = kh * 8;
    for (int vi = 0; vi < 8; ++vi) {
        const int mm = vi + mr;
        dst[(size_t)(row0 + mm) * UDIM + n0]      = acc0[vi];
        dst[(size_t)(row0 + mm) * UDIM + n0 + 16] = acc1[vi];
    }
}

// ---------------------------------------------------------------------------
// Async copy helpers (gfx1250 GLOBAL_LOAD_ASYNC_TO_LDS_*, tracked by ASYNCcnt).
// The LDS-address operand is the wave-relative LDS byte offset; an AS(3)->flat
// addrspacecast puts that offset in the low 32 bits of the generic pointer.
// ---------------------------------------------------------------------------
__device__ __forceinline__ void async_copy_b128(void* lds_dst, const void* gsrc) {
    unsigned int  la = (unsigned int)(unsigned long long)lds_dst;   // LDS byte offset
    unsigned long long ga = (unsigned long long)gsrc;
    asm volatile("global_load_async_to_lds_b128 %0, %1, off"
                 :: "v"(la), "v"(ga)
                 : "memory");
}
__device__ __forceinline__ void wait_async_le2() {
    asm volatile("s_wait_asynccnt 0x2" ::: "memory");
}
__device__ __forceinline__ void wait_async_0() {
    asm volatile("s_wait_asynccnt 0x0" ::: "memory");
}

// ---------------------------------------------------------------------------
// Kernel 2: fused per-edge-type transform.  Each block processes
// TILES_PER_BLK consecutive 16-edge tiles with a double-buffered LDS A tile
// filled by async-to-LDS copies (next tile streams in while WMMAs run).
//   gate_e = ef @ W_gate_e[t];  val_e = ef @ W_dense[t] + b_dense[t]
//   gate   = sigmoid(src_gated[idx] + gate_e)
//   atomically scatter-add val_e * gate into out[idx]
// ---------------------------------------------------------------------------
#define EPITCH 132          // 128 + 4 pad (row stride 528B: 16B aligned, banks spread)
#define TILES_PER_BLK 8
__global__ __launch_bounds__(256) void edge_fused(
    const float* __restrict__ edge_feats,  // [T, E, FE]
    const int*   __restrict__ edge_idx,    // [T, E]
    const float* __restrict__ W_gate_e,    // [T, FE, U]
    const float* __restrict__ W_dense,     // [T, FE, U]
    const float* __restrict__ b_dense,     // [T, U]
    const float* __restrict__ src_gated,   // [BN, U]  (workspace)
    float* __restrict__ out)               // [BN, U]
{
    __shared__ float As[2][16 * EPITCH];

    const int blocksPerType = (NEDGE / 16) / TILES_PER_BLK;   // 1024
    const int t     = blockIdx.x / blocksPerType;
    const int tile0 = (blockIdx.x % blocksPerType) * TILES_PER_BLK;

    const float* ef_t  = edge_feats + (size_t)t * NEDGE * FEDIM;
    const int*   idx_t = edge_idx   + (size_t)t * NEDGE;
    const float* Wgt   = W_gate_e   + (size_t)t * FEDIM * UDIM;
    const float* Wdt   = W_dense    + (size_t)t * FEDIM * UDIM;

    const int lane = threadIdx.x & 31;
    const int wave = threadIdx.x >> 5;
    const int m    = lane & 15;
    const int kh   = lane >> 4;
    const int n0   = wave * 32 + m;

    const float bb0 = b_dense[(size_t)t * UDIM + n0];
    const float bb1 = b_dense[(size_t)t * UDIM + n0 + 16];

    // This thread's two 16B chunks of each 16x128 tile (512 b128 transfers/tile)
    const int i0 = threadIdx.x;            // chunk 0
    const int i1 = threadIdx.x + 256;      // chunk 1
    const int r0 = i0 >> 5, c0 = (i0 & 31) * 4;
    const int r1 = i1 >> 5, c1 = (i1 & 31) * 4;

    // Prologue: kick off async fill of tile0 into buffer 0
    {
        const float* src = ef_t + (size_t)(tile0 * 16) * FEDIM;
        async_copy_b128(&As[0][r0 * EPITCH + c0], src + (size_t)r0 * FEDIM + c0);
        async_copy_b128(&As[0][r1 * EPITCH + c1], src + (size_t)r1 * FEDIM + c1);
    }

    for (int j = 0; j < TILES_PER_BLK; ++j) {
        const int buf = j & 1;

        if (j + 1 < TILES_PER_BLK) {
            // Stream next tile into the other buffer while we compute this one
            const float* src = ef_t + (size_t)((tile0 + j + 1) * 16) * FEDIM;
            async_copy_b128(&As[buf ^ 1][r0 * EPITCH + c0], src + (size_t)r0 * FEDIM + c0);
            async_copy_b128(&As[buf ^ 1][r1 * EPITCH + c1], src + (size_t)r1 * FEDIM + c1);
            wait_async_le2();   // retire the older (current-tile) pair
        } else {
            wait_async_0();
        }
        __syncthreads();        // all waves' fills of As[buf] are complete

        const float* at = As[buf];

        v8f ag0 = {}, ag1 = {};   // gate accumulators (two 16-col tiles)
        v8f av0 = {}, av1 = {};   // value accumulators

        for (int k = 0; k < FEDIM; k += 4) {
            const int kk = k + 2 * kh;
            v2f a = *(const v2f*)&at[m * EPITCH + kk];
            v2f bg0, bg1, bv0, bv1;
            bg0.x = Wgt[(size_t)kk * UDIM + n0];
            bg0.y = Wgt[(size_t)(kk + 1) * UDIM + n0];
            bg1.x = Wgt[(size_t)kk * UDIM + n0 + 16];
            bg1.y = Wgt[(size_t)(kk + 1) * UDIM + n0 + 16];
            bv0.x = Wdt[(size_t)kk * UDIM + n0];
            bv0.y = Wdt[(size_t)(kk + 1) * UDIM + n0];
            bv1.x = Wdt[(size_t)kk * UDIM + n0 + 16];
            bv1.y = Wdt[(size_t)(kk + 1) * UDIM + n0 + 16];
            ag0 = __builtin_amdgcn_wmma_f32_16x16x4_f32(false, a, false, bg0,
                                                        (short)0, ag0, false, false);
            ag1 = __builtin_amdgcn_wmma_f32_16x16x4_f32(false, a, false, bg1,
                                                        (short)0, ag1, false, false);
            av0 = __builtin_amdgcn_wmma_f32_16x16x4_f32(false, a, false, bv0,
                                                        (short)0, av0, false, false);
            av1 = __builtin_amdgcn_wmma_f32_16x16x4_f32(false, a, false, bv1,
                                                        (short)0, av1, false, false);
        }

        // Epilogue: C layout (lane->N, vgpr->M) gives coalesced 64B gathers and
        // 64B-contiguous f32 atomics for the scatter-add.
        const int e0 = (tile0 + j) * 16;
        const int mr = kh * 8;
        for (int vi = 0; vi < 8; ++vi) {
            const int mm  = vi + mr;
            const int row = idx_t[e0 + mm];          // broadcast per half-wave, L2 hit
            const size_t o0 = (size_t)row * UDIM + n0;
            const float sg0 = src_gated[o0];
            const float sg1 = src_gated[o0 + 16];
            const float gate0 = 1.f / (1.f + __expf(-(ag0[vi] + sg0)));
            const float gate1 = 1.f / (1.f + __expf(-(ag1[vi] + sg1)));
            const float msg0 = (av0[vi] + bb0) * gate0;
            const float msg1 = (av1[vi] + bb1) * gate1;
            unsafeAtomicAdd(&out[o0],      msg0);    // -> global_atomic_add_f32
            unsafeAtomicAdd(&out[o0 + 16], msg1);
        }
        __syncthreads();   // everyone done reading As[buf] before it is refilled
    }
}

// ---------------------------------------------------------------------------
extern "C" void kernel_launch(void* const* d_in, const int* in_sizes, int n_in,
                              void* d_out, int out_size, void* d_ws, size_t ws_size,
                              hipStream_t stream) {
    const float* features   = (const float*)d_in[0];  // [B,N,F]
    const int*   edge_idx   = (const int*)  d_in[1];  // [T,E]
    const float* edge_feats = (const float*)d_in[2];  // [T,E,FE]
    const float* W_gate     = (const float*)d_in[3];  // [F,U]
    const float* W_gate_e   = (const float*)d_in[4];  // [T,FE,U]
    const float* W_dense    = (const float*)d_in[5];  // [T,FE,U]
    const float* b_dense    = (const float*)d_in[6];  // [T,U]
    float*       out        = (float*)d_out;          // [B,N,U]
    float*       src_gated  = (float*)d_ws;           // [BN,U] scratch (102.4 MB)

    (void)in_sizes; (void)n_in; (void)out_size; (void)ws_size;

    // 1) zero output accumulator
    zero_f32<<<2048, 256, 0, stream>>>(out, (BN * UDIM) / 4);

    // 2) shared node-gate GEMM into workspace
    gate_gemm<<<BN / 16, 256, 0, stream>>>(features, W_gate, src_gated);

    // 3) fused per-edge-type GEMMs + gather + sigmoid-gate + scatter-add
    //    (double-buffered async-to-LDS tile pipeline, 8 tiles per block)
    edge_fused<<<NTYPE * ((NEDGE / 16) / TILES_PER_BLK), 256, 0, stream>>>(
        edge_feats, edge_idx, W_gate_e, W_dense, b_dense, src_gated, out);
}